// Lucas_Kanade_layer_11261404250306
// MI455X (gfx1250) — compile-verified
//
#include <hip/hip_runtime.h>
#include <hip/hip_bf16.h>

typedef __attribute__((ext_vector_type(2))) float v2f;
typedef __attribute__((ext_vector_type(8))) float v8f;

#define BB    4
#define CC    128
#define TH_   160
#define TW_   160
#define TN    (TH_*TW_)     /* 25600 */
#define IH_   192
#define IW_   192
#define IN_   (IH_*IW_)     /* 36864 */
#define NWG   800           /* one wave per WG; 800*32 = 25600 points per batch */
#define EPS_DIAG 1e-4f

__device__ __forceinline__ int clampi(int v, int lo, int hi) {
  return v < lo ? lo : (v > hi ? hi : v);
}

// ---------------------------------------------------------------------------
// Kernel 0: zero the 4 x 16 x 16 accumulation scratch
// ---------------------------------------------------------------------------
__global__ void lk_zero(float* __restrict__ scr) {
  int i = blockIdx.x * 256 + threadIdx.x;
  if (i < BB * 256) scr[i] = 0.f;
}

// ---------------------------------------------------------------------------
// Kernel 1: per-point moments -> rank-2 Cholesky rows -> WMMA Gram accumulate
// One wave (32 points) per workgroup: 3200 independent waves fill the WGPs,
// and workgroup barriers degenerate to S_NOP (single-wave workgroup).
// ---------------------------------------------------------------------------
__global__ __launch_bounds__(32) void lk_accum(
    const float* __restrict__ tfeat, const float* __restrict__ ifeat,
    const float* __restrict__ twgt,  const float* __restrict__ iwgt,
    const float* __restrict__ mat,   float* __restrict__ scr)
{
  __shared__ float zbuf[32][2][16];   // [point][z-row][component]
  const int b    = blockIdx.y;
  const int wg   = blockIdx.x;        // wave id in batch [0, 800)
  const int lane = threadIdx.x & 31;

  const float M00 = mat[b*9+0], M01 = mat[b*9+1], M02 = mat[b*9+2];
  const float M10 = mat[b*9+3], M11 = mat[b*9+4], M12 = mat[b*9+5];
  const float M20 = mat[b*9+6], M21 = mat[b*9+7], M22 = mat[b*9+8];

  const float* tb = tfeat + (size_t)b * CC * TN;
  const float* ib = ifeat + (size_t)b * CC * IN_;
  const float* wb = iwgt  + (size_t)b * IN_;
  const float* vb = twgt  + (size_t)b * TN;

  v8f acc = {0.f,0.f,0.f,0.f,0.f,0.f,0.f,0.f};

  const int n  = wg * 32 + lane;      // point index [0, 25600)
  const int ix = n % TW_;
  const int iy = n / TW_;
  const float x = (float)ix, y = (float)iy;

  // ---- homography warp ----
  float tpx = M00*x + M01*y + M02;
  float tpy = M10*x + M11*y + M12;
  float tpz = M20*x + M21*y + M22;
  tpz = (fabsf(tpz) > 1e-8f) ? tpz : 1e-8f;
  const float px = tpx / tpz, py = tpy / tpz;

  const float fx0 = floorf(px), fy0 = floorf(py);
  const float fx = px - fx0,  fy = py - fy0;
  const int x0i = clampi((int)fx0,     0, IW_-1);
  const int x1i = clampi((int)fx0 + 1, 0, IW_-1);
  const int y0i = clampi((int)fy0,     0, IH_-1);
  const int y1i = clampi((int)fy0 + 1, 0, IH_-1);
  const float w00 = (1.f-fx)*(1.f-fy);
  const float w10 = fx*(1.f-fy);
  const float w01 = (1.f-fx)*fy;
  const float w11 = fx*fy;
  const int i00 = y0i*IW_ + x0i;
  const int i10 = y0i*IW_ + x1i;
  const int i01 = y1i*IW_ + x0i;
  const int i11 = y1i*IW_ + x1i;
  const bool inb = (px >= 0.f) && (px <= (float)(IW_-1)) &&
                   (py >= 0.f) && (py <= (float)(IH_-1));

  const float w2 = w00*wb[i00] + w10*wb[i10] + w01*wb[i01] + w11*wb[i11];
  const float wn = inb ? (vb[n] * w2) : 0.f;

  // Reference J pairs gradients via reshape(b,-1,C) WITHOUT transpose:
  // du'[n,c] = du_flat[n*C + c] over (c,h,w) flat order.
  // Since 128*200 == 25600: plane = n/200, spatial base = (n%200)*128.
  const int   ca     = n / 200;
  const int   sbase  = (n % 200) * CC;
  const float* gpl   = tb + (size_t)ca * TN;

  // Incremental strip indexing: no per-iteration div/mod.
  int xa   = sbase % TW_;
  int ya   = sbase / TW_;
  int rowc = ya * TW_;

  float ma = 0.f, mb = 0.f, md = 0.f, gu = 0.f, gv = 0.f;
  #pragma unroll 2
  for (int c = 0; c < CC; ++c) {
    const int xm = (xa == 0)     ? 1         : xa - 1;   // reflect pad
    const int xp = (xa == TW_-1) ? TW_ - 2   : xa + 1;
    const int rm = (ya == 0)     ? TW_       : rowc - TW_;
    const int rp = (ya == TH_-1) ? (TH_-2)*TW_ : rowc + TW_;
    const float du = gpl[rowc + xp] - gpl[rowc + xm];
    const float dv = gpl[rp + xa]  - gpl[rm + xa];

    const float* cp = ib + (size_t)c * IN_;
    __builtin_prefetch(cp + IN_ + i00, 0, 0);            // next channel plane
    const float interp = w00*cp[i00] + w10*cp[i10] + w01*cp[i01] + w11*cp[i11];
    const float res = tb[(size_t)c * TN + n] - interp;   // res = -(interp - templ)

    ma = fmaf(du, du, ma);
    mb = fmaf(du, dv, mb);
    md = fmaf(dv, dv, md);
    gu = fmaf(du, res, gu);
    gv = fmaf(dv, res, gv);

    ++xa;
    if (xa == TW_) { xa = 0; ++ya; rowc += TW_; }
  }

  ma *= wn; mb *= wn; md *= wn; gu *= wn; gv *= wn;

  // 2x2 Cholesky of [[ma,mb],[mb,md]]; augmented column (index 8) carries grad.
  const float l11   = sqrtf(fmaxf(ma, 0.f));
  const float inv11 = (l11 > 1e-20f) ? 1.f / l11 : 0.f;
  const float l21   = mb * inv11;
  const float l22   = sqrtf(fmaxf(md - l21*l21, 0.f));
  const float inv22 = (l22 > 1e-20f) ? 1.f / l22 : 0.f;
  const float alpha = gu * inv11;
  const float beta  = (gv - alpha * l21) * inv22;

  const float r1[8] = {x,   y,   1.f, 1.f, 1.f, 1.f, -x*x, -x*y};
  const float r2[8] = {1.f, 1.f, 1.f, x,   y,   1.f, -x*y, -y*y};

  float* z0 = &zbuf[lane][0][0];
  float* z1 = &zbuf[lane][1][0];
  #pragma unroll
  for (int i = 0; i < 8; ++i) {
    z0[i] = fmaf(l11, r1[i], l21 * r2[i]);
    z1[i] = l22 * r2[i];
  }
  z0[8] = alpha; z1[8] = beta;
  #pragma unroll
  for (int i = 9; i < 16; ++i) { z0[i] = 0.f; z1[i] = 0.f; }

  __syncthreads();   // single-wave workgroup: S_NOP; DS in-order per wave

  // D += Z^T Z via V_WMMA_F32_16X16X4_F32.
  // A 16x4 layout: lanes 0-15 v0/v1 = K0/K1, lanes 16-31 v0/v1 = K2/K3.
  // K rows 0..3 = {z0(pt 2t), z1(pt 2t), z0(pt 2t+1), z1(pt 2t+1)};
  // identical lane data serves the B (4x16) operand.
  #pragma unroll
  for (int t = 0; t < 16; ++t) {
    const int pp   = 2*t + (lane >> 4);
    const int comp = lane & 15;
    v2f abv;
    abv.x = zbuf[pp][0][comp];
    abv.y = zbuf[pp][1][comp];
    acc = __builtin_amdgcn_wmma_f32_16x16x4_f32(false, abv, false, abv,
                                                (short)0, acc, false, false);
  }

  // Only the 9x9 top-left block of D is consumed: Hess = D[0:8][0:8],
  // grad = D[0:8][8].  VGPR r of lanes 0-15 holds D[M=r][N=lane].
  if (lane < 9) {
    float* sbp = scr + b * 256;
    #pragma unroll
    for (int r = 0; r < 8; ++r)
      atomicAdd(&sbp[r * 16 + lane], acc[r]);
  }
}

// ---------------------------------------------------------------------------
// Kernel 2: per-batch  H = Hess/N + eps*I ; delta = -H^{-1} grad (8x8 Cholesky)
// ---------------------------------------------------------------------------
__global__ void lk_solve(const float* __restrict__ scr, float* __restrict__ out) {
  const int b = threadIdx.x;
  if (b >= BB) return;
  const float invN = 1.0f / (float)TN;
  float Hm[8][8], g[8];
  for (int i = 0; i < 8; ++i) {
    g[i] = scr[b*256 + i*16 + 8] * invN;
    for (int j = 0; j < 8; ++j) Hm[i][j] = scr[b*256 + i*16 + j] * invN;
  }
  for (int i = 0; i < 8; ++i) Hm[i][i] += EPS_DIAG;

  float L[8][8];
  for (int k = 0; k < 8; ++k) {
    float s = Hm[k][k];
    for (int m = 0; m < k; ++m) s -= L[k][m]*L[k][m];
    L[k][k] = sqrtf(fmaxf(s, 1e-30f));
    const float inv = 1.f / L[k][k];
    for (int i = k+1; i < 8; ++i) {
      float t = Hm[i][k];
      for (int m = 0; m < k; ++m) t -= L[i][m]*L[k][m];
      L[i][k] = t * inv;
    }
  }
  float yv[8];
  for (int i = 0; i < 8; ++i) {
    float s = g[i];
    for (int m = 0; m < i; ++m) s -= L[i][m]*yv[m];
    yv[i] = s / L[i][i];
  }
  float xv[8];
  for (int i = 7; i >= 0; --i) {
    float s = yv[i];
    for (int m = i+1; m < 8; ++m) s -= L[m][i]*xv[m];
    xv[i] = s / L[i][i];
  }
  for (int i = 0; i < 8; ++i) out[b*8 + i] = -xv[i];
}

// ---------------------------------------------------------------------------
extern "C" void kernel_launch(void* const* d_in, const int* in_sizes, int n_in,
                              void* d_out, int out_size, void* d_ws, size_t ws_size,
                              hipStream_t stream) {
  (void)in_sizes; (void)n_in; (void)out_size; (void)ws_size;
  const float* tfeat = (const float*)d_in[0];   // (4,128,160,160)
  const float* ifeat = (const float*)d_in[1];   // (4,128,192,192)
  const float* twgt  = (const float*)d_in[2];   // (4,1,160,160)
  const float* iwgt  = (const float*)d_in[3];   // (4,1,192,192)
  const float* mat   = (const float*)d_in[4];   // (4,3,3)
  float* out = (float*)d_out;                   // (4,8,1)
  float* scr = (float*)d_ws;                    // 4*256 floats

  lk_zero<<<4, 256, 0, stream>>>(scr);
  dim3 grid(NWG, BB);
  lk_accum<<<grid, 32, 0, stream>>>(tfeat, ifeat, twgt, iwgt, mat, scr);
  lk_solve<<<1, 32, 0, stream>>>(scr, out);
}